// HGCN_65274912964681
// MI455X (gfx1250) — compile-verified
//
#include <hip/hip_runtime.h>

#define N_NODES 100000
#define N_EDGES 1600000
#define FEAT    128
#define NBRANCH 5

typedef __attribute__((ext_vector_type(2))) float v2f;
typedef __attribute__((ext_vector_type(8))) float v8f;

// ---------------- utility kernels ----------------

__global__ void k_fill(float* __restrict__ p, float v, int n) {
    int i = blockIdx.x * blockDim.x + threadIdx.x;
    if (i < n) p[i] = v;
}

// edge_index int64 (2,E) -> int32 src/dst
__global__ void k_cvt_edges(const long long* __restrict__ ei,
                            int* __restrict__ src, int* __restrict__ dst) {
    int e = blockIdx.x * blockDim.x + threadIdx.x;
    if (e < N_EDGES) {
        src[e] = (int)ei[e];
        dst[e] = (int)ei[N_EDGES + e];
    }
}

// deg[i][n] = 1 + sum over edges of mask_i at dst  (deg pre-filled with 1.0)
__global__ void k_deg(const int* __restrict__ dst,
                      const long long* __restrict__ etype,
                      float* __restrict__ deg) {
    int e = blockIdx.x * blockDim.x + threadIdx.x;
    if (e < N_EDGES) {
        int d = dst[e];
#pragma unroll
        for (int i = 0; i < 4; ++i) {
            if ((int)etype[(long long)e * 4 + i] == 1)
                atomicAdd(&deg[i * N_NODES + d], 1.0f);
        }
        atomicAdd(&deg[4 * N_NODES + d], 1.0f);
    }
}

// dinv = deg^-0.5 ; dinv2 = 1/deg  (deg >= 1 always)
__global__ void k_dinv(const float* __restrict__ deg,
                       float* __restrict__ dinv, float* __restrict__ dinv2) {
    int i = blockIdx.x * blockDim.x + threadIdx.x;
    if (i < NBRANCH * N_NODES) {
        float dv = rsqrtf(deg[i]);
        dinv[i]  = dv;
        dinv2[i] = dv * dv;
    }
}

// coef[i][e] = dinv_i[src]*dinv_i[dst]*mask_i(e)
__global__ void k_coef(const int* __restrict__ src, const int* __restrict__ dst,
                       const long long* __restrict__ etype,
                       const float* __restrict__ dinv,
                       float* __restrict__ coef) {
    int e = blockIdx.x * blockDim.x + threadIdx.x;
    if (e < N_EDGES) {
        int s = src[e], d = dst[e];
#pragma unroll
        for (int i = 0; i < 4; ++i) {
            float m = ((int)etype[(long long)e * 4 + i] == 1) ? 1.0f : 0.0f;
            coef[i * N_EDGES + e] = m * dinv[i * N_NODES + s] * dinv[i * N_NODES + d];
        }
        coef[4 * N_EDGES + e] = dinv[4 * N_NODES + s] * dinv[4 * N_NODES + d];
    }
}

// agg[n][f] = x[n][f] * dinv2_i[n]   (self-loop message; also initializes agg)
__global__ void k_agg_init(const float* __restrict__ x,
                           const float* __restrict__ dinv2_i,
                           float* __restrict__ agg) {
    int idx = blockIdx.x * blockDim.x + threadIdx.x;
    if (idx < N_NODES * FEAT) {
        int n = idx >> 7;
        agg[idx] = x[idx] * dinv2_i[n];
    }
}

// one wave32 per edge: gather x[src] row (float4/lane) and atomically add
// coef * row into agg[dst]. coef==0 -> whole wave skips (uniform branch).
__global__ void k_scatter(const float* __restrict__ x,
                          const int* __restrict__ src, const int* __restrict__ dst,
                          const float* __restrict__ coef_i,
                          float* __restrict__ agg) {
    int gw   = (blockIdx.x * blockDim.x + threadIdx.x) >> 5;
    int lane = threadIdx.x & 31;
    if (gw >= N_EDGES) return;
    float c = coef_i[gw];
    if (c == 0.0f) return;
    int s = src[gw], d = dst[gw];
    const float4 v = ((const float4*)(x + (size_t)s * FEAT))[lane];
    float* ag = agg + (size_t)d * FEAT + lane * 4;
    atomicAdd(ag + 0, c * v.x);
    atomicAdd(ag + 1, c * v.y);
    atomicAdd(ag + 2, c * v.z);
    atomicAdd(ag + 3, c * v.w);
}

// out[16-row tile] += 0.2 * relu(agg_tile @ W + b)
// 256 threads = 8 waves; wave w owns the 16x16 tile at columns [16w,16w+16).
// A tile (16x128 f32) staged in LDS with pitch 132 (4-dword skew -> the 16
// row-reads per half-wave land on distinct banks).
__global__ void __launch_bounds__(256)
k_gemm_acc(const float* __restrict__ agg, const float* __restrict__ W,
           const float* __restrict__ bias, float* __restrict__ out) {
    __shared__ float sA[16 * 132];
    const int row0 = blockIdx.x * 16;
    const int t = threadIdx.x;

    // coalesced A-tile load: thread t -> row t/16, 8 consecutive floats
    {
        int r = t >> 4;
        int cb = (t & 15) * 8;
        const float* sp = agg + (size_t)(row0 + r) * FEAT + cb;
        float* dp = sA + r * 132 + cb;
#pragma unroll
        for (int q = 0; q < 8; ++q) dp[q] = sp[q];
    }
    __syncthreads();

    const int wave = t >> 5;
    const int lane = t & 31;
    const int hi   = lane >> 4;   // 0: K pair (k,k+1), 1: (k+2,k+3)
    const int lo   = lane & 15;
    const int c0   = wave * 16;

    v8f acc = {};
#pragma unroll 4
    for (int k = 0; k < FEAT; k += 4) {
        int kk = k + 2 * hi;
        v2f a, b;
        a.x = sA[lo * 132 + kk];
        a.y = sA[lo * 132 + kk + 1];
        b.x = W[kk * FEAT + c0 + lo];
        b.y = W[(kk + 1) * FEAT + c0 + lo];
        acc = __builtin_amdgcn_wmma_f32_16x16x4_f32(
            /*neg_a=*/false, a, /*neg_b=*/false, b,
            /*c_mod=*/(short)0, acc, /*reuse_a=*/false, /*reuse_b=*/false);
    }

    const float bv = bias[c0 + lo];
    float* op = out + (size_t)row0 * FEAT;
#pragma unroll
    for (int j = 0; j < 8; ++j) {
        int grow = j + 8 * hi;           // C/D layout: VGPR j -> M=j (+8 for hi lanes)
        int gcol = c0 + lo;
        float v = fmaxf(acc[j] + bv, 0.0f) * 0.2f;
        float* p = op + grow * FEAT + gcol;
        *p = *p + v;
    }
}

// ---------------- host launcher ----------------

extern "C" void kernel_launch(void* const* d_in, const int* in_sizes, int n_in,
                              void* d_out, int out_size, void* d_ws, size_t ws_size,
                              hipStream_t stream) {
    (void)in_sizes; (void)n_in; (void)out_size; (void)ws_size;

    const float*     x  = (const float*)d_in[0];
    const long long* ei = (const long long*)d_in[1];
    const long long* et = (const long long*)d_in[2];
    const float*     W1 = (const float*)d_in[3];
    const float*     b1 = (const float*)d_in[4];
    const float*     W2 = (const float*)d_in[5];
    const float*     b2 = (const float*)d_in[6];
    float* out = (float*)d_out;

    // workspace layout
    char* ws = (char*)d_ws;
    int*   src32 = (int*)ws;              ws += (size_t)N_EDGES * 4;
    int*   dst32 = (int*)ws;              ws += (size_t)N_EDGES * 4;
    float* deg   = (float*)ws;            ws += (size_t)NBRANCH * N_NODES * 4;
    float* dinv  = (float*)ws;            ws += (size_t)NBRANCH * N_NODES * 4;
    float* dinv2 = (float*)ws;            ws += (size_t)NBRANCH * N_NODES * 4;
    float* coef  = (float*)ws;            ws += (size_t)NBRANCH * N_EDGES * 4;
    float* h1    = (float*)ws;            ws += (size_t)N_NODES * FEAT * 4;
    float* agg   = (float*)ws;            ws += (size_t)N_NODES * FEAT * 4;

    const int TPB = 256;
    const int gE  = (N_EDGES + TPB - 1) / TPB;
    const int g5N = (NBRANCH * N_NODES + TPB - 1) / TPB;
    const int gNF = (N_NODES * FEAT + TPB - 1) / TPB;
    const int gScat = N_EDGES / 8;             // 8 waves per 256-thread block
    const int gGemm = N_NODES / 16;            // 100000 % 16 == 0

    // graph-only precompute (shared by both layers)
    k_cvt_edges<<<gE, TPB, 0, stream>>>(ei, src32, dst32);
    k_fill<<<g5N, TPB, 0, stream>>>(deg, 1.0f, NBRANCH * N_NODES);
    k_deg<<<gE, TPB, 0, stream>>>(dst32, et, deg);
    k_dinv<<<g5N, TPB, 0, stream>>>(deg, dinv, dinv2);
    k_coef<<<gE, TPB, 0, stream>>>(src32, dst32, et, dinv, coef);

    // layer 1: x -> h1
    k_fill<<<gNF, TPB, 0, stream>>>(h1, 0.0f, N_NODES * FEAT);
    for (int i = 0; i < NBRANCH; ++i) {
        k_agg_init<<<gNF, TPB, 0, stream>>>(x, dinv2 + i * N_NODES, agg);
        k_scatter<<<gScat, TPB, 0, stream>>>(x, src32, dst32, coef + (size_t)i * N_EDGES, agg);
        k_gemm_acc<<<gGemm, TPB, 0, stream>>>(agg, W1 + (size_t)i * FEAT * FEAT,
                                              b1 + (size_t)i * FEAT, h1);
    }

    // layer 2: h1 -> out
    k_fill<<<gNF, TPB, 0, stream>>>(out, 0.0f, N_NODES * FEAT);
    for (int i = 0; i < NBRANCH; ++i) {
        k_agg_init<<<gNF, TPB, 0, stream>>>(h1, dinv2 + i * N_NODES, agg);
        k_scatter<<<gScat, TPB, 0, stream>>>(h1, src32, dst32, coef + (size_t)i * N_EDGES, agg);
        k_gemm_acc<<<gGemm, TPB, 0, stream>>>(agg, W2 + (size_t)i * FEAT * FEAT,
                                              b2 + (size_t)i * FEAT, out);
    }
}